// minGRU_29137058136562
// MI455X (gfx1250) — compile-verified
//
#include <hip/hip_runtime.h>
#include <math.h>
#include <stdint.h>

typedef __attribute__((ext_vector_type(2))) float v2f;
typedef __attribute__((ext_vector_type(8))) float v8f;
typedef __attribute__((ext_vector_type(4))) unsigned int u32x4;
typedef __attribute__((ext_vector_type(4))) int i32x4;
typedef __attribute__((ext_vector_type(8))) int i32x8;

constexpr int Bb = 8;
constexpr int Ss = 4096;
constexpr int Dd = 1024;
constexpr int Mtot = Bb * Ss;          // 32768 rows
constexpr int NC = 32;                 // scan chunks per channel
constexpr int CL = Ss / NC;            // 128 steps per chunk
constexpr int KC = 32;                 // K-chunk staged in LDS
constexpr int NKC = Dd / KC;           // 32 K-chunks
constexpr int LDSS = 36;               // padded LDS row stride (floats)

#if __has_builtin(__builtin_amdgcn_tensor_load_to_lds)
#define USE_TDM 1
#else
#define USE_TDM 0
#endif

__device__ __forceinline__ float softplus_f(float x) {
    return fmaxf(x, 0.0f) + log1pf(expf(-fabsf(x)));
}
__device__ __forceinline__ float log_g_f(float x) {
    return (x >= 0.0f) ? logf(x + 0.5f) : -softplus_f(-x);
}
__device__ __forceinline__ float logaddexp_f(float a, float b) {
    if (a == -INFINITY) return b;
    if (b == -INFINITY) return a;
    float m = fmaxf(a, b);
    return m + log1pf(expf(-fabsf(a - b)));
}

// gfx1250 async DMA: global -> LDS, 16B per lane, tracked by ASYNCcnt.
// VDST supplies the per-lane LDS byte offset (HW adds wave LDS_BASE); the low
// 32 bits of a generic pointer to __shared__ are exactly that offset.
__device__ __forceinline__ void async_b128(const float* gptr, const float* lptr) {
    uint32_t lds = (uint32_t)(uintptr_t)lptr;
    uint64_t ga  = (uint64_t)(uintptr_t)gptr;
    asm volatile("global_load_async_to_lds_b128 %0, %1, off"
                 :: "v"(lds), "v"(ga) : "memory");
}
__device__ __forceinline__ void wait_async0() {
    asm volatile("s_wait_asynccnt 0x0" ::: "memory");
}

#if USE_TDM
// Tensor Data Mover: DMA a 2D fp32 tile (tile_d1 rows x tile_d0 elems, row
// stride `stride_el` elements) from global into LDS, inserting 4 DWORDs of
// LDS padding after every 32 DWORDs stored (=> 36-float padded rows for a
// 32-float tile row). Tracked by TENSORcnt. Issue from ONE wave only.
__device__ __forceinline__ void tdm_load_2d(const float* gptr, const float* ldsptr,
                                            unsigned tile_d0, unsigned tile_d1,
                                            unsigned stride_el) {
    uint64_t ga = (uint64_t)(uintptr_t)gptr;
    u32x4 g0;
    g0[0] = 1u;                                            // count=1 (valid, user)
    g0[1] = (uint32_t)(uintptr_t)ldsptr;                   // lds_addr (bytes)
    g0[2] = (uint32_t)(ga & 0xFFFFFFFFu);                  // global_addr[31:0]
    g0[3] = (uint32_t)((ga >> 32) & 0x01FFFFFFu)           // global_addr[56:32]
          | (2u << 30);                                    // type = 2 (image)
    i32x8 g1;
    g1[0] = (int)((2u << 16)      // data_size = 4 bytes
          | (1u << 20)            // pad_enable
          | (4u << 22)            // pad_interval: 32 DWORDs
          | (3u << 25));          // pad_amount:   4 DWORDs
    g1[1] = (int)(tile_d0 << 16);              // tensor_dim0[15:0] = tile_d0
    g1[2] = (int)(tile_d1 << 16);              // tensor_dim1[15:0] = tile_d1
    g1[3] = (int)(tile_d0 << 16);              // tile_dim0
    g1[4] = (int)(tile_d1 & 0xFFFFu);          // tile_dim1 (tile_dim2 = 0)
    g1[5] = (int)stride_el;                    // tensor_dim0_stride[31:0]
    g1[6] = 0;
    g1[7] = 0;
    i32x4 z4; z4[0] = 0; z4[1] = 0; z4[2] = 0; z4[3] = 0;  // groups 2/3 unused (2D)
    i32x8 z8;
#pragma unroll
    for (int i = 0; i < 8; ++i) z8[i] = 0;
    __builtin_amdgcn_tensor_load_to_lds(g0, g1, z4, z4, z8, 0);
}
__device__ __forceinline__ void wait_tensor0() {
    __builtin_amdgcn_s_wait_tensorcnt(0);
}
#endif

// ---------------------------------------------------------------------------
// Stage 1: fused dual GEMM (k = x W_z^T, u = x W_h^T) + log-space epilogue.
//   lc_out[m*D+n] = -softplus(k)                    (log(1 - z))
//   lv_out[m*D+n] = -softplus(-k) + log_g(u)        (log z + log g(~h))
// Tile: 128 rows x 64 cols per block; 8 wave32s. x tile via TDM (wave 0),
// weight tiles via per-thread async-to-LDS; double-buffered.
// ---------------------------------------------------------------------------
__global__ __launch_bounds__(256) void fused_dual_gemm(
    const float* __restrict__ x, const float* __restrict__ Wz,
    const float* __restrict__ Wh, float* __restrict__ lc_out,
    float* __restrict__ lv_out)
{
    __shared__ float xs[2][128 * LDSS];
    __shared__ float zs[2][64 * LDSS];
    __shared__ float hs[2][64 * LDSS];

    const int tid  = threadIdx.x;
    const int lane = tid & 31;
    const int wv   = tid >> 5;
    const int m0   = blockIdx.y * 128;
    const int n0   = blockIdx.x * 64;

    const int wr = tid >> 3;           // weight row slot (x2 via i-loop)
    const int c4 = (tid & 7) * 4;      // float4 column within K-chunk
#if !USE_TDM
    const int xr = tid >> 3;           // x row slot (x4 via i-loop)
#endif

    auto issue_weights = [&](int buf, int k0) {
#pragma unroll
        for (int i = 0; i < 2; ++i) {
            int r = wr + i * 32;
            size_t g = (size_t)(n0 + r) * Dd + k0 + c4;
            async_b128(Wz + g, &zs[buf][r * LDSS + c4]);
            async_b128(Wh + g, &hs[buf][r * LDSS + c4]);
        }
    };
    auto issue_x = [&](int buf, int k0) {
#if USE_TDM
        if (wv == 0)
            tdm_load_2d(x + (size_t)m0 * Dd + k0, &xs[buf][0], KC, 128, Dd);
#else
#pragma unroll
        for (int i = 0; i < 4; ++i) {
            int r = xr + i * 32;
            async_b128(x + (size_t)(m0 + r) * Dd + k0 + c4,
                       &xs[buf][r * LDSS + c4]);
        }
#endif
    };

    v8f acck[4], accu[4];
#pragma unroll
    for (int i = 0; i < 4; ++i)
#pragma unroll
        for (int j = 0; j < 8; ++j) { acck[i][j] = 0.0f; accu[i][j] = 0.0f; }

    const int arow = (wv * 16 + (lane & 15)) * LDSS;
    const int koff = (lane >> 4) * 2;

    issue_weights(0, 0);
    issue_x(0, 0);

    for (int kc = 0; kc < NKC; ++kc) {
        const int cur = kc & 1;
        wait_async0();            // my async (weight) loads landed
#if USE_TDM
        if (wv == 0) wait_tensor0();   // the x-tile DMA landed
#endif
        __syncthreads();          // all shares landed; prior reads retired
        if (kc + 1 < NKC) {
            issue_weights(1 - cur, (kc + 1) * KC);  // overwrite retired buffer
            issue_x(1 - cur, (kc + 1) * KC);
        }

        const float* __restrict__ xsb = xs[cur];
        const float* __restrict__ zsb = zs[cur];
        const float* __restrict__ hsb = hs[cur];

#pragma unroll
        for (int kk = 0; kk < KC; kk += 4) {
            // A fragment: 16x4 fp32 (rows across lanes 0-15, K split by lane half)
            v2f a;
            a.x = xsb[arow + kk + koff];
            a.y = xsb[arow + kk + koff + 1];
#pragma unroll
            for (int nt = 0; nt < 4; ++nt) {
                int brow = (nt * 16 + (lane & 15)) * LDSS + kk + koff;
                v2f bz; bz.x = zsb[brow]; bz.y = zsb[brow + 1];
                v2f bh; bh.x = hsb[brow]; bh.y = hsb[brow + 1];
                acck[nt] = __builtin_amdgcn_wmma_f32_16x16x4_f32(
                    false, a, false, bz, (short)0, acck[nt], false, false);
                accu[nt] = __builtin_amdgcn_wmma_f32_16x16x4_f32(
                    false, a, false, bh, (short)0, accu[nt], false, false);
            }
        }
    }

    // Epilogue: C/D layout -> n = lane&15, m = 8*(lane>>4) + v
    const int ncol  = lane & 15;
    const int mbase = (lane >> 4) * 8;
#pragma unroll
    for (int nt = 0; nt < 4; ++nt) {
#pragma unroll
        for (int v = 0; v < 8; ++v) {
            int m = m0 + wv * 16 + mbase + v;
            int n = n0 + nt * 16 + ncol;
            size_t idx = (size_t)m * Dd + n;
            float kv = acck[nt][v];
            float uv = accu[nt][v];
            lc_out[idx] = -softplus_f(kv);
            lv_out[idx] = -softplus_f(-kv) + log_g_f(uv);
        }
    }
}

// ---------------------------------------------------------------------------
// Stage 2a: per-chunk reduction. Thread = (b, chunk, d). Computes
//   sumA = sum(lc) over chunk, r = fold logaddexp(lc_i + r, lv_i) from -inf.
// ---------------------------------------------------------------------------
__global__ __launch_bounds__(256) void scan_chunk_reduce(
    const float* __restrict__ lc, const float* __restrict__ lv,
    float* __restrict__ sumA, float* __restrict__ rloc)
{
    int t = blockIdx.x * blockDim.x + threadIdx.x;     // B*NC*D threads
    int d = t % Dd;
    int c = (t / Dd) % NC;
    int b = t / (Dd * NC);
    size_t base = ((size_t)b * Ss + (size_t)c * CL) * Dd + d;

    float sA = 0.0f;
    float r  = -INFINITY;
    for (int i = 0; i < CL; ++i) {
        size_t idx = base + (size_t)i * Dd;
        if (i + 16 < CL) {
            __builtin_prefetch(lc + idx + 16 * Dd, 0, 1);
            __builtin_prefetch(lv + idx + 16 * Dd, 0, 1);
        }
        float a  = lc[idx];
        float vv = lv[idx];
        sA += a;
        r = logaddexp_f(r + a, vv);
    }
    sumA[t] = sA;
    rloc[t] = r;
}

// ---------------------------------------------------------------------------
// Stage 2b: sequential combine across chunks per channel (tiny, L2-resident).
// carry[b,c,d] = incoming log-state for chunk c.
// ---------------------------------------------------------------------------
__global__ __launch_bounds__(256) void scan_carry(
    const float* __restrict__ h_prev, const float* __restrict__ sumA,
    const float* __restrict__ rloc, float* __restrict__ carry)
{
    int t = blockIdx.x * blockDim.x + threadIdx.x;     // B*D threads
    int d = t % Dd;
    int b = t / Dd;
    float y = log_g_f(h_prev[(size_t)b * Dd + d]);
    for (int c = 0; c < NC; ++c) {
        size_t j = ((size_t)b * NC + c) * Dd + d;
        carry[j] = y;
        y = logaddexp_f(y + sumA[j], rloc[j]);
    }
}

// ---------------------------------------------------------------------------
// Stage 2c: replay chunk with true carry; out holds lc on entry, h on exit.
// ---------------------------------------------------------------------------
__global__ __launch_bounds__(256) void scan_apply(
    const float* __restrict__ lv, const float* __restrict__ carry,
    float* __restrict__ out)
{
    int t = blockIdx.x * blockDim.x + threadIdx.x;     // B*NC*D threads
    int d = t % Dd;
    int c = (t / Dd) % NC;
    int b = t / (Dd * NC);
    size_t base = ((size_t)b * Ss + (size_t)c * CL) * Dd + d;

    float y = carry[t];
    for (int i = 0; i < CL; ++i) {
        size_t idx = base + (size_t)i * Dd;
        if (i + 16 < CL) {
            __builtin_prefetch(out + idx + 16 * Dd, 0, 1);
            __builtin_prefetch(lv + idx + 16 * Dd, 0, 1);
        }
        float a  = out[idx];     // lc, read before overwrite
        float vv = lv[idx];
        y = logaddexp_f(y + a, vv);
        out[idx] = expf(y);
    }
}

// ---------------------------------------------------------------------------
extern "C" void kernel_launch(void* const* d_in, const int* in_sizes, int n_in,
                              void* d_out, int out_size, void* d_ws, size_t ws_size,
                              hipStream_t stream) {
    const float* x      = (const float*)d_in[0];
    const float* h_prev = (const float*)d_in[1];
    const float* Wz     = (const float*)d_in[2];
    const float* Wh     = (const float*)d_in[3];
    float* out = (float*)d_out;

    size_t n_el = (size_t)Bb * Ss * Dd;
    float* lv    = (float*)d_ws;                 // B*S*D
    float* sumA  = lv + n_el;                    // B*NC*D
    float* rloc  = sumA + (size_t)Bb * NC * Dd;  // B*NC*D
    float* carry = rloc + (size_t)Bb * NC * Dd;  // B*NC*D

    dim3 g1(Dd / 64, Mtot / 128);
    fused_dual_gemm<<<g1, 256, 0, stream>>>(x, Wz, Wh, out, lv);

    int t2 = Bb * NC * Dd;
    scan_chunk_reduce<<<t2 / 256, 256, 0, stream>>>(out, lv, sumA, rloc);
    scan_carry<<<(Bb * Dd) / 256, 256, 0, stream>>>(h_prev, sumA, rloc, carry);
    scan_apply<<<t2 / 256, 256, 0, stream>>>(lv, carry, out);
}